// EigenvectorAttentionModel_67181878444766
// MI455X (gfx1250) — compile-verified
//
#include <hip/hip_runtime.h>
#include <hip/hip_bf16.h>
#include <math.h>
#include <stdint.h>

typedef __attribute__((ext_vector_type(2))) float v2f;
typedef __attribute__((ext_vector_type(8))) float v8f;

#define B_  8
#define T_  176
#define IND_ 116
#define H_  128
#define NH_ 8
#define EPS_ 1e-5f
#define CORR_CH 44   // K-chunk staged in LDS for corr kernel (176 = 4*44)

__device__ __forceinline__ float sigmf(float x) { return 1.f / (1.f + __expf(-x)); }

__device__ __forceinline__ v8f wmma4(v2f a, v2f b, v8f c) {
    return __builtin_amdgcn_wmma_f32_16x16x4_f32(false, a, false, b, (short)0, c, false, false);
}

// ---------------------------------------------------------------------------
// Generic batched WMMA GEMM: C = act(alpha * A @ op(B) + bias)
//  Each wave computes a 16x32 strip (2 N-tiles share the A fragment, dual
//  K-accumulators break the WMMA dependency chain). Branch-free main loop:
//  OOB rows/cols are clamped for loads and masked only at the store.
//  A: M x K (lda); B: N x K if TB else K x N (ldb); C: M x N (ldc).
//  batch z = z1*nb2+z2 adds z1*s?1 + z2*s?2 element offsets.
// ---------------------------------------------------------------------------
template <int TB>
__global__ void gemm_wmma_t(const float* __restrict__ A, const float* __restrict__ B,
                            const float* __restrict__ bias, float* __restrict__ C,
                            int M, int N, int K, int lda, int ldb, int ldc,
                            int act, float alpha, int nb2,
                            long sA1, long sA2, long sB1, long sB2, long sC1, long sC2) {
    int lane = threadIdx.x & 31;
    int wave = threadIdx.x >> 5;
    int tn = (blockIdx.x * 4 + wave) * 32;
    int tm = blockIdx.y * 16;
    if (tm >= M || tn >= N) return;
    int z = blockIdx.z, z1 = z / nb2, z2 = z % nb2;
    A += z1 * sA1 + z2 * sA2;
    B += z1 * sB1 + z2 * sB2;
    C += z1 * sC1 + z2 * sC2;

    int hl = lane & 15;
    int khalf = (lane >> 4) << 1;
    int row  = tm + hl;
    int rowc = min(row, M - 1);
    int col0 = tn + hl;
    int col1 = col0 + 16;
    int col0c = min(col0, N - 1);
    int col1c = min(col1, N - 1);

    const float* Ar = A + (long)rowc * lda;
    const float* B0;
    const float* B1;
    if (TB) { B0 = B + (long)col0c * ldb; B1 = B + (long)col1c * ldb; }
    else    { B0 = B + col0c;             B1 = B + col1c; }

    v8f acc0a = {}, acc0b = {}, acc1a = {}, acc1b = {};
    int K8 = K & ~7;
    int k0 = 0;
    for (; k0 < K8; k0 += 8) {
        v2f aA = *(const v2f*)(Ar + k0 + khalf);
        v2f aB = *(const v2f*)(Ar + k0 + 4 + khalf);
        v2f b0A, b0B, b1A, b1B;
        if (TB) {
            b0A = *(const v2f*)(B0 + k0 + khalf);
            b0B = *(const v2f*)(B0 + k0 + 4 + khalf);
            b1A = *(const v2f*)(B1 + k0 + khalf);
            b1B = *(const v2f*)(B1 + k0 + 4 + khalf);
        } else {
            long r0 = (long)(k0 + khalf) * ldb;
            long r1 = (long)(k0 + 4 + khalf) * ldb;
            b0A.x = B0[r0]; b0A.y = B0[r0 + ldb];
            b1A.x = B1[r0]; b1A.y = B1[r0 + ldb];
            b0B.x = B0[r1]; b0B.y = B0[r1 + ldb];
            b1B.x = B1[r1]; b1B.y = B1[r1 + ldb];
        }
        acc0a = wmma4(aA, b0A, acc0a);
        acc1a = wmma4(aA, b1A, acc1a);
        acc0b = wmma4(aB, b0B, acc0b);
        acc1b = wmma4(aB, b1B, acc1b);
    }
    for (; k0 < K; k0 += 4) {  // tail (K not multiple of 8): clamped loads + select
        int ka = k0 + khalf, kb = ka + 1;
        int kac = min(ka, K - 1), kbc = min(kb, K - 1);
        v2f a, b0, b1;
        float t;
        t = Ar[kac]; a.x = (ka < K) ? t : 0.f;
        t = Ar[kbc]; a.y = (kb < K) ? t : 0.f;
        if (TB) {
            t = B0[kac]; b0.x = (ka < K) ? t : 0.f;
            t = B0[kbc]; b0.y = (kb < K) ? t : 0.f;
            t = B1[kac]; b1.x = (ka < K) ? t : 0.f;
            t = B1[kbc]; b1.y = (kb < K) ? t : 0.f;
        } else {
            t = B0[(long)kac * ldb]; b0.x = (ka < K) ? t : 0.f;
            t = B0[(long)kbc * ldb]; b0.y = (kb < K) ? t : 0.f;
            t = B1[(long)kac * ldb]; b1.x = (ka < K) ? t : 0.f;
            t = B1[(long)kbc * ldb]; b1.y = (kb < K) ? t : 0.f;
        }
        acc0a = wmma4(a, b0, acc0a);
        acc1a = wmma4(a, b1, acc1a);
    }

    v8f s0 = acc0a + acc0b;
    v8f s1 = acc1a + acc1b;
    int cr = tm + ((lane >> 4) << 3);
    #pragma unroll
    for (int r = 0; r < 8; ++r) {
        int rr = cr + r;
        if (rr < M && col0 < N) {
            float v = s0[r] * alpha;
            if (bias) v += bias[col0];
            if (act == 1) v = fmaxf(v, 0.f);
            else if (act == 2) v = sigmf(v);
            C[(long)rr * ldc + col0] = v;
        }
        if (rr < M && col1 < N) {
            float v = s1[r] * alpha;
            if (bias) v += bias[col1];
            if (act == 1) v = fmaxf(v, 0.f);
            else if (act == 2) v = sigmf(v);
            C[(long)rr * ldc + col1] = v;
        }
    }
}

// ---------------------------------------------------------------------------
// gmpool l1: rows are pairwise distances dist[b,i,j,:] computed on the fly.
// C[row, n] = sum_k sqrt((Xe[b,i,k]-Xe[b,j,k])^2+1e-9) * W[n,k] + bias[n]
// Xe layout: row (i*B + b), H columns. K = Hh (multiple of 8), branch-free.
// ---------------------------------------------------------------------------
__global__ void dist_gemm_wmma(const float* __restrict__ Xe, const float* __restrict__ W,
                               const float* __restrict__ bias, float* __restrict__ C,
                               int Tt, int Bb, int Hh) {
    int lane = threadIdx.x & 31;
    int wave = threadIdx.x >> 5;
    int tn = (blockIdx.x * 4 + wave) * 32;
    int tm = blockIdx.y * 16;
    int M = Bb * Tt * Tt;
    if (tm >= M || tn >= Hh) return;

    int hl = lane & 15;
    int khalf = (lane >> 4) << 1;
    int row = tm + hl;
    int rowc = min(row, M - 1);
    int b = rowc / (Tt * Tt);
    int rem = rowc % (Tt * Tt);
    int ii = rem / Tt, jj = rem % Tt;
    const float* xi = Xe + ((long)(ii * Bb + b)) * Hh;
    const float* xj = Xe + ((long)(jj * Bb + b)) * Hh;
    int col0 = tn + hl;
    int col1 = col0 + 16;
    const float* B0 = W + (long)col0 * Hh;
    const float* B1 = W + (long)col1 * Hh;

    v8f acc0a = {}, acc0b = {}, acc1a = {}, acc1b = {};
    for (int k0 = 0; k0 < Hh; k0 += 8) {
        v2f piA = *(const v2f*)(xi + k0 + khalf);
        v2f pjA = *(const v2f*)(xj + k0 + khalf);
        v2f piB = *(const v2f*)(xi + k0 + 4 + khalf);
        v2f pjB = *(const v2f*)(xj + k0 + 4 + khalf);
        v2f aA, aB;
        float d;
        d = piA.x - pjA.x; aA.x = __fsqrt_rn(d * d + 1e-9f);
        d = piA.y - pjA.y; aA.y = __fsqrt_rn(d * d + 1e-9f);
        d = piB.x - pjB.x; aB.x = __fsqrt_rn(d * d + 1e-9f);
        d = piB.y - pjB.y; aB.y = __fsqrt_rn(d * d + 1e-9f);
        v2f b0A = *(const v2f*)(B0 + k0 + khalf);
        v2f b0B = *(const v2f*)(B0 + k0 + 4 + khalf);
        v2f b1A = *(const v2f*)(B1 + k0 + khalf);
        v2f b1B = *(const v2f*)(B1 + k0 + 4 + khalf);
        acc0a = wmma4(aA, b0A, acc0a);
        acc1a = wmma4(aA, b1A, acc1a);
        acc0b = wmma4(aB, b0B, acc0b);
        acc1b = wmma4(aB, b1B, acc1b);
    }
    v8f s0 = acc0a + acc0b;
    v8f s1 = acc1a + acc1b;
    int cr = tm + ((lane >> 4) << 3);
    #pragma unroll
    for (int r = 0; r < 8; ++r) {
        int rr = cr + r;
        if (rr < M) {
            C[(long)rr * Hh + col0] = s0[r] + bias[col0];
            C[(long)rr * Hh + col1] = s1[r] + bias[col1];
        }
    }
}

// ---------------------------------------------------------------------------
// Weighted covariance per (b,t): y[h,s]=S[b,s,t]*Xe[b,s,h]; Cov=(y-m)(y-m)^T/(T-1)
// One block (256 thr = 8 waves) per (b,t). Raw Xe chunks are DMA'd into LDS
// with CDNA5 async loads (global_load_async_to_lds_b128, ASYNCcnt), then all
// 8 waves consume them; the w^2 weight is folded into the A fragment since
// G[h,g] = sum_s w_s^2 Xe[s,h] Xe[s,g].
// ---------------------------------------------------------------------------
__global__ void corr_wmma(const float* __restrict__ Xe, const float* __restrict__ Smat,
                          float* __restrict__ Cov, int Tt, int Bb, int Hh) {
    __shared__ float wS[T_];
    __shared__ float mS[H_];
    __shared__ __align__(16) float Zs[CORR_CH * H_];
    int bt = blockIdx.x;
    int bb = bt / Tt, t = bt % Tt;
    int tid = threadIdx.x;
    if (tid < Tt) wS[tid] = Smat[(long)bb * Tt * Tt + (long)tid * Tt + t];
    __syncthreads();
    if (tid < Hh) {
        float s = 0.f;
        for (int k = 0; k < Tt; ++k) s += wS[k] * Xe[((long)(k * Bb + bb)) * Hh + tid];
        mS[tid] = s / Tt;
    }

    int lane = tid & 31;
    int wave = tid >> 5;
    int hl = lane & 15;
    int khalf = (lane >> 4) << 1;
    int rowh = wave * 16 + hl;
    unsigned ldsbase = (unsigned)(uintptr_t)&Zs[0];  // low 32 bits = LDS offset

    v8f acc[8];
    #pragma unroll
    for (int g = 0; g < 8; ++g) acc[g] = (v8f){};

    for (int kc = 0; kc < Tt; kc += CORR_CH) {
        __syncthreads();  // previous chunk fully consumed
        // async DMA: rows [kc, kc+CORR_CH) of Xe (for this b) -> Zs
        for (int i4 = tid; i4 < CORR_CH * Hh / 4; i4 += 256) {
            int e = i4 * 4;
            int s = kc + e / Hh;
            int h = e % Hh;
            const float* gp = Xe + ((long)(s * Bb + bb)) * Hh + h;
            unsigned loff = ldsbase + (unsigned)(i4 * 16);
            asm volatile("global_load_async_to_lds_b128 %0, %1, off"
                         :: "v"(loff), "v"(gp) : "memory");
        }
        asm volatile("s_wait_asynccnt 0" ::: "memory");
        __syncthreads();
        for (int k0 = 0; k0 < CORR_CH; k0 += 4) {
            int ka = (k0 + khalf) * Hh;
            float wa = wS[kc + k0 + khalf];
            float wb = wS[kc + k0 + khalf + 1];
            v2f a;
            a.x = wa * wa * Zs[ka + rowh];
            a.y = wb * wb * Zs[ka + Hh + rowh];
            #pragma unroll
            for (int g = 0; g < 8; ++g) {
                v2f b;
                b.x = Zs[ka + g * 16 + hl];
                b.y = Zs[ka + Hh + g * 16 + hl];
                acc[g] = wmma4(a, b, acc[g]);
            }
        }
    }

    long base = (long)bt * Hh * Hh;
    int cr = wave * 16 + ((lane >> 4) << 3);
    #pragma unroll
    for (int g = 0; g < 8; ++g) {
        int cc = g * 16 + hl;
        #pragma unroll
        for (int r = 0; r < 8; ++r) {
            int rr = cr + r;
            float v = (acc[g][r] - (float)Tt * mS[rr] * mS[cc]) / (float)(Tt - 1);
            Cov[base + (long)rr * Hh + cc] = v;
        }
    }
}

__global__ void corr_normalize(float* __restrict__ Cov, int Hh) {
    __shared__ float stdS[H_];
    long base = (long)blockIdx.x * Hh * Hh;
    int tid = threadIdx.x;
    if (tid < Hh) stdS[tid] = __fsqrt_rn(fmaxf(Cov[base + (long)tid * Hh + tid], 1e-30f));
    __syncthreads();
    for (int i = tid; i < Hh * Hh; i += blockDim.x) {
        int h = i / Hh, g = i % Hh;
        float r = Cov[base + i] / (stdS[h] * stdS[g]);
        Cov[base + i] = fminf(1.f, fmaxf(-1.f, r));
    }
}

// ---------------------------------------------------------------------------
// BatchNorm over rows (N == 128 in all uses): coalesced 3-phase mean/var.
// ---------------------------------------------------------------------------
__global__ void bn_zero(float* __restrict__ mv, int N) {
    int i = blockIdx.x * blockDim.x + threadIdx.x;
    if (i < 2 * N) mv[i] = 0.f;
}

__global__ void bn_acc(const float* __restrict__ X, float* __restrict__ mv, long M, int N) {
    int c = threadIdx.x;  // N threads, one per column -> coalesced row reads
    float s = 0.f, s2 = 0.f;
    for (long r = blockIdx.x; r < M; r += gridDim.x) {
        float v = X[r * N + c];
        s += v; s2 += v * v;
    }
    atomicAdd(&mv[c], s);
    atomicAdd(&mv[N + c], s2);
}

__global__ void bn_finalize(float* __restrict__ mv, long M, int N) {
    int c = threadIdx.x;
    if (c < N) {
        float m = mv[c] / (float)M;
        mv[c] = m;
        mv[N + c] = mv[N + c] / (float)M - m * m;
    }
}

__global__ void bn_apply(float* __restrict__ X, const float* __restrict__ mv,
                         const float* __restrict__ g, const float* __restrict__ bb,
                         long M, int N, int relu) {
    long total = M * N;
    for (long i = (long)blockIdx.x * blockDim.x + threadIdx.x; i < total;
         i += (long)gridDim.x * blockDim.x) {
        int c = (int)(i % N);
        float y = g[c] * (X[i] - mv[c]) * rsqrtf(mv[N + c] + EPS_) + bb[c];
        if (relu) y = fmaxf(y, 0.f);
        X[i] = y;
    }
}

// ---------------------------------------------------------------------------
// LayerNorm over last dim (optional residual R added first).
// ---------------------------------------------------------------------------
__global__ void layernorm_rows(const float* __restrict__ X, const float* __restrict__ R,
                               const float* __restrict__ g, const float* __restrict__ bb,
                               float* __restrict__ Y, int N) {
    __shared__ float red[256];
    __shared__ float mS, vS;
    long row = blockIdx.x;
    int tid = threadIdx.x;
    const float* xr = X + row * N;
    const float* rr = R ? R + row * N : nullptr;
    float s = 0.f;
    for (int c = tid; c < N; c += 256) s += xr[c] + (rr ? rr[c] : 0.f);
    red[tid] = s; __syncthreads();
    for (int st = 128; st > 0; st >>= 1) { if (tid < st) red[tid] += red[tid + st]; __syncthreads(); }
    if (tid == 0) mS = red[0] / N;
    __syncthreads();
    float m = mS; s = 0.f;
    for (int c = tid; c < N; c += 256) {
        float v = xr[c] + (rr ? rr[c] : 0.f) - m;
        s += v * v;
    }
    red[tid] = s; __syncthreads();
    for (int st = 128; st > 0; st >>= 1) { if (tid < st) red[tid] += red[tid + st]; __syncthreads(); }
    if (tid == 0) vS = red[0] / N;
    __syncthreads();
    float inv = rsqrtf(vS + EPS_);
    for (int c = tid; c < N; c += 256) {
        float v = xr[c] + (rr ? rr[c] : 0.f);
        Y[row * N + c] = g[c] * (v - m) * inv + bb[c];
    }
}

__global__ void softmax_rows(float* __restrict__ X, int N) {
    __shared__ float red[256];
    __shared__ float mS, sS;
    long row = blockIdx.x;
    int tid = threadIdx.x;
    float* xr = X + row * N;
    float mx = -1e30f;
    for (int c = tid; c < N; c += 256) mx = fmaxf(mx, xr[c]);
    red[tid] = mx; __syncthreads();
    for (int st = 128; st > 0; st >>= 1) { if (tid < st) red[tid] = fmaxf(red[tid], red[tid + st]); __syncthreads(); }
    if (tid == 0) mS = red[0];
    __syncthreads();
    float s = 0.f;
    for (int c = tid; c < N; c += 256) { float e = __expf(xr[c] - mS); xr[c] = e; s += e; }
    red[tid] = s; __syncthreads();
    for (int st = 128; st > 0; st >>= 1) { if (tid < st) red[tid] += red[tid + st]; __syncthreads(); }
    if (tid == 0) sS = red[0];
    __syncthreads();
    float inv = 1.f / sS;
    for (int c = tid; c < N; c += 256) xr[c] *= inv;
}

// ---------------------------------------------------------------------------
// LSTM scan (T steps, B=8, H=128). xW = x@Wih^T + bih precomputed.
// ---------------------------------------------------------------------------
__global__ void lstm_kernel(const float* __restrict__ xW, const float* __restrict__ whh,
                            const float* __restrict__ bhh, float* __restrict__ xout,
                            int Tt, int Bb, int Hh) {
    __shared__ float hS[B_ * H_], cS[B_ * H_], gS[B_ * 4 * H_];
    int tid = threadIdx.x;  // 512
    for (int i = tid; i < Bb * Hh; i += blockDim.x) { hS[i] = 0.f; cS[i] = 0.f; }
    __syncthreads();
    int G = 4 * Hh;
    for (int t = 0; t < Tt; ++t) {
        for (int j = tid; j < G; j += blockDim.x) {
            const float4* wr4 = (const float4*)(whh + (long)j * Hh);
            float bj = bhh[j];
            for (int b = 0; b < Bb; ++b) {
                const float4* hr4 = (const float4*)(hS + b * Hh);
                float acc = xW[((long)(b * Tt + t)) * G + j] + bj;
                for (int k = 0; k < Hh / 4; ++k) {
                    float4 w4 = wr4[k];
                    float4 h4 = hr4[k];
                    acc += w4.x * h4.x + w4.y * h4.y + w4.z * h4.z + w4.w * h4.w;
                }
                gS[b * G + j] = acc;
            }
        }
        __syncthreads();
        for (int idx = tid; idx < Bb * Hh; idx += blockDim.x) {
            int b = idx / Hh, k = idx % Hh;
            float gi = sigmf(gS[b * G + k]);
            float gf = sigmf(gS[b * G + Hh + k]);
            float gg = tanhf(gS[b * G + 2 * Hh + k]);
            float go = sigmf(gS[b * G + 3 * Hh + k]);
            float c = gf * cS[idx] + gi * gg;
            float h = go * tanhf(c);
            cS[idx] = c; hS[idx] = h;
            xout[((long)(t * Bb + b)) * Hh + k] = h;
        }
        __syncthreads();
    }
}

// ---------------------------------------------------------------------------
// clf -> M matrix (diag=1), V = I
// ---------------------------------------------------------------------------
__global__ void build_M_initV(const float* __restrict__ clf, float* __restrict__ A,
                              float* __restrict__ V, int Tt, int Bb) {
    long total = (long)Bb * Tt * Tt;
    for (long i = (long)blockIdx.x * blockDim.x + threadIdx.x; i < total;
         i += (long)gridDim.x * blockDim.x) {
        long rem = i % ((long)Tt * Tt);
        int ii = (int)(rem / Tt), jj = (int)(rem % Tt);
        A[i] = (ii == jj) ? 1.f : clf[i];
        V[i] = (ii == jj) ? 1.f : 0.f;
    }
}

// ---------------------------------------------------------------------------
// Parallel two-sided Jacobi eigensolver: one block per matrix.
// ---------------------------------------------------------------------------
__global__ void jacobi_kernel(float* __restrict__ A, float* __restrict__ V, int n, int sweeps) {
    __shared__ float cS[96], sS[96];
    __shared__ int pS[96], qS[96];
    int b = blockIdx.x, tid = threadIdx.x;
    float* Ab = A + (long)b * n * n;
    float* Vb = V + (long)b * n * n;
    int m = n / 2;
    for (int sw = 0; sw < sweeps; ++sw) {
        for (int r = 0; r < n - 1; ++r) {
            if (tid < m) {
                int k = tid, p, q;
                if (k == 0) { p = n - 1; q = r % (n - 1); }
                else { p = (r + k) % (n - 1); q = (r - k + (n - 1)) % (n - 1); }
                float app = Ab[(long)p * n + p], aqq = Ab[(long)q * n + q], apq = Ab[(long)p * n + q];
                float c = 1.f, s = 0.f;
                if (fabsf(apq) > 1e-12f) {
                    float tau = (aqq - app) / (2.f * apq);
                    float tt = (tau >= 0.f ? 1.f : -1.f) / (fabsf(tau) + __fsqrt_rn(1.f + tau * tau));
                    c = rsqrtf(1.f + tt * tt);
                    s = tt * c;
                }
                cS[k] = c; sS[k] = s; pS[k] = p; qS[k] = q;
            }
            __syncthreads();
            for (int task = tid; task < m * n; task += blockDim.x) {
                int pr = task / n, j = task % n;
                int p = pS[pr], q = qS[pr];
                float c = cS[pr], s = sS[pr];
                float ap = Ab[(long)p * n + j], aq = Ab[(long)q * n + j];
                Ab[(long)p * n + j] = c * ap - s * aq;
                Ab[(long)q * n + j] = s * ap + c * aq;
            }
            __syncthreads();
            for (int task = tid; task < m * n; task += blockDim.x) {
                int pr = task / n, i = task % n;
                int p = pS[pr], q = qS[pr];
                float c = cS[pr], s = sS[pr];
                float ap = Ab[(long)i * n + p], aq = Ab[(long)i * n + q];
                Ab[(long)i * n + p] = c * ap - s * aq;
                Ab[(long)i * n + q] = s * ap + c * aq;
                float vp = Vb[(long)i * n + p], vq = Vb[(long)i * n + q];
                Vb[(long)i * n + p] = c * vp - s * vq;
                Vb[(long)i * n + q] = s * vp + c * vq;
            }
            __syncthreads();
        }
    }
}

__global__ void eigsort_scale(const float* __restrict__ A, const float* __restrict__ V,
                              float* __restrict__ S, int n) {
    __shared__ float vals[T_];
    __shared__ int ord[T_];
    __shared__ float sq[T_];
    int b = blockIdx.x, tid = threadIdx.x;
    if (tid < n) { vals[tid] = A[((long)b * n + tid) * n + tid]; ord[tid] = tid; }
    __syncthreads();
    if (tid == 0) {
        for (int i = 0; i < n - 1; ++i) {
            int best = i;
            for (int j = i + 1; j < n; ++j) if (vals[j] > vals[best]) best = j;
            float tv = vals[i]; vals[i] = vals[best]; vals[best] = tv;
            int to = ord[i]; ord[i] = ord[best]; ord[best] = to;
        }
    }
    __syncthreads();
    if (tid < n) {
        float v = vals[tid];
        if (v <= 0.f) v = 1e-7f;
        sq[tid] = __fsqrt_rn(v);
    }
    __syncthreads();
    for (int idx = tid; idx < n * n; idx += blockDim.x) {
        int i = idx / n, j = idx % n;
        S[(long)b * n * n + idx] = V[((long)b * n + i) * n + ord[j]] * sq[j];
    }
}

// ---------------------------------------------------------------------------
// misc elementwise / reduction kernels
// ---------------------------------------------------------------------------
__global__ void concat_vt(const float* __restrict__ v, const float* __restrict__ t,
                          float* __restrict__ h, long rows, int Dv, int Dt) {
    int D = Dv + Dt;
    long total = rows * D;
    for (long i = (long)blockIdx.x * blockDim.x + threadIdx.x; i < total;
         i += (long)gridDim.x * blockDim.x) {
        long row = i / D; int c = (int)(i % D);
        h[i] = (c < Dv) ? v[row * Dv + c] : t[row * Dt + (c - Dv)];
    }
}

__global__ void concat_es(const float* __restrict__ fcseq, const float* __restrict__ es,
                          float* __restrict__ out, int Tt, int Hh) {
    long total = (long)B_ * Tt * 2 * Hh;
    for (long i = (long)blockIdx.x * blockDim.x + threadIdx.x; i < total;
         i += (long)gridDim.x * blockDim.x) {
        long row = i / (2 * Hh); int c = (int)(i % (2 * Hh));
        int b = (int)(row / Tt);
        out[i] = (c < Hh) ? fcseq[row * Hh + c] : es[(long)b * Hh + (c - Hh)];
    }
}

__global__ void mean_over_t(const float* __restrict__ Hn, float* __restrict__ out, int Tt, int Hh) {
    int b = blockIdx.x, h = threadIdx.x;
    float s = 0.f;
    for (int t = 0; t < Tt; ++t) s += Hn[((long)(b * Tt + t)) * Hh + h];
    out[(long)b * Hh + h] = s / Tt;
}

__global__ void garo_att_ev(const float* __restrict__ xq, const float* __restrict__ xk,
                            const float* __restrict__ Hn, float* __restrict__ ev,
                            int Tt, int Hh) {
    __shared__ float red[H_];
    __shared__ float attS;
    int blk = blockIdx.x;
    int b = blk / Tt, t = blk % Tt;
    int tid = threadIdx.x;  // Hh threads
    long r = (long)(b * Tt + t) * Hh;
    red[tid] = xq[(long)b * Hh + tid] * xk[r + tid];
    __syncthreads();
    for (int st = Hh / 2; st > 0; st >>= 1) { if (tid < st) red[tid] += red[tid + st]; __syncthreads(); }
    if (tid == 0) attS = sigmf(red[0] / __fsqrt_rn((float)Hh));
    __syncthreads();
    ev[r + tid] = Hn[r + tid] * attS;
}

__global__ void sum_over_l(const float* __restrict__ ha, float* __restrict__ hdyn, int L, int Bb, int E) {
    int b = blockIdx.x, s = threadIdx.x;
    if (s >= E) return;
    float acc = 0.f;
    for (int l = 0; l < L; ++l) acc += ha[((long)(l * Bb + b)) * E + s];
    hdyn[(long)b * E + s] = acc;
}

// ---------------------------------------------------------------------------
// Host-side helpers
// ---------------------------------------------------------------------------
static inline void launch_gemm(hipStream_t s, const float* A, const float* B, const float* bias,
                               float* C, int M, int N, int K, int lda, int ldb, int ldc,
                               int transB, int act, float alpha,
                               int nb1 = 1, int nb2 = 1,
                               long sA1 = 0, long sA2 = 0, long sB1 = 0, long sB2 = 0,
                               long sC1 = 0, long sC2 = 0) {
    dim3 grid((N + 127) / 128, (M + 15) / 16, nb1 * nb2);
    if (transB)
        gemm_wmma_t<1><<<grid, 128, 0, s>>>(A, B, bias, C, M, N, K, lda, ldb, ldc,
                                            act, alpha, nb2, sA1, sA2, sB1, sB2, sC1, sC2);
    else
        gemm_wmma_t<0><<<grid, 128, 0, s>>>(A, B, bias, C, M, N, K, lda, ldb, ldc,
                                            act, alpha, nb2, sA1, sA2, sB1, sB2, sC1, sC2);
}

static void run_bn(hipStream_t s, float* X, float* mv, long M, int N,
                   const float* g, const float* bb, int relu) {
    bn_zero<<<1, 256, 0, s>>>(mv, N);
    int blocks = (int)min((long)512, M);
    bn_acc<<<blocks, N, 0, s>>>(X, mv, M, N);
    bn_finalize<<<1, N, 0, s>>>(mv, M, N);
    long tot = M * N;
    bn_apply<<<(unsigned)((tot + 255) / 256), 256, 0, s>>>(X, mv, g, bb, M, N, relu);
}

static void run_mlp2(hipStream_t s, const float* X, long M, int Kin, int Hh,
                     const float* l1w, const float* l1b, const float* n1g, const float* n1b,
                     const float* l2w, const float* l2b, const float* n2g, const float* n2b,
                     float* tmp, float* out, float* mv) {
    launch_gemm(s, X, l1w, l1b, tmp, (int)M, Hh, Kin, Kin, Kin, Hh, 1, 0, 1.f);
    run_bn(s, tmp, mv, M, Hh, n1g, n1b, 1);
    launch_gemm(s, tmp, l2w, l2b, out, (int)M, Hh, Hh, Hh, Hh, Hh, 1, 0, 1.f);
    run_bn(s, out, mv, M, Hh, n2g, n2b, 1);
}

static void run_transformer(hipStream_t s, const float* x2d, int E, int hid, int nh,
                            const float* inw, const float* inb, const float* outw, const float* outb,
                            const float* ln1g, const float* ln1b, const float* ln2g, const float* ln2b,
                            const float* m1w, const float* m1b, const float* m2w, const float* m2b,
                            float* qkv, float* scores, float* obuf, float* xa, float* hidtmp,
                            float* tmpd, float* out) {
    const int L = T_, Bb = B_;
    const int rows = L * Bb;
    const int dh = E / nh;
    const int E3 = 3 * E;
    // qkv projection
    launch_gemm(s, x2d, inw, inb, qkv, rows, E3, E, E, E, E3, 1, 0, 1.f);
    // scores[b,h] = Q K^T / sqrt(dh)
    launch_gemm(s, qkv, qkv + E, nullptr, scores, L, L, dh,
                Bb * E3, Bb * E3, L, 1, 0, 1.f / sqrtf((float)dh),
                Bb, nh, E3, dh, E3, dh, (long)nh * L * L, (long)L * L);
    softmax_rows<<<Bb * nh * L, 256, 0, s>>>(scores, L);
    // o = att @ V
    launch_gemm(s, scores, qkv + 2 * E, nullptr, obuf, L, dh, L,
                L, Bb * E3, Bb * E, 0, 0, 1.f,
                Bb, nh, (long)nh * L * L, (long)L * L, E3, dh, E, dh);
    // out projection -> tmpd, then LN1 -> xa
    launch_gemm(s, obuf, outw, outb, tmpd, rows, E, E, E, E, E, 1, 0, 1.f);
    layernorm_rows<<<rows, 256, 0, s>>>(tmpd, nullptr, ln1g, ln1b, xa, E);
    // FFN
    launch_gemm(s, xa, m1w, m1b, hidtmp, rows, hid, E, E, E, hid, 1, 1, 1.f);
    launch_gemm(s, hidtmp, m2w, m2b, tmpd, rows, E, hid, hid, hid, E, 1, 0, 1.f);
    layernorm_rows<<<rows, 256, 0, s>>>(xa, tmpd, ln2g, ln2b, out, E);
}

// ---------------------------------------------------------------------------
extern "C" void kernel_launch(void* const* d_in, const int* in_sizes, int n_in,
                              void* d_out, int out_size, void* d_ws, size_t ws_size,
                              hipStream_t stream) {
    (void)in_sizes; (void)n_in; (void)out_size; (void)ws_size;
    #define IN(i) ((const float*)d_in[(i)])
    // input leaf indices (setup_inputs dict order, params flattened recursively)
    const float* V   = IN(0);
    const float* Tm  = IN(1);
    const float* FC  = IN(2);
    // se mlp2
    const float *se_l1w=IN(3), *se_l1b=IN(4), *se_n1g=IN(5), *se_n1b=IN(6);
    const float *se_l2w=IN(7), *se_l2b=IN(8), *se_n2g=IN(9), *se_n2b=IN(10);
    // mlp mlp2
    const float *mp_l1w=IN(11), *mp_l1b=IN(12), *mp_n1g=IN(13), *mp_n1b=IN(14);
    const float *mp_l2w=IN(15), *mp_l2b=IN(16), *mp_n2g=IN(17), *mp_n2b=IN(18);
    // lstm
    const float *wih=IN(19), *whh=IN(20), *bih=IN(21), *bhh=IN(22);
    // tatt
    const float *t1_inw=IN(23), *t1_inb=IN(24), *t1_outw=IN(25), *t1_outb=IN(26);
    const float *t1_ln1g=IN(27), *t1_ln1b=IN(28), *t1_ln2g=IN(29), *t1_ln2b=IN(30);
    const float *t1_m1w=IN(31), *t1_m1b=IN(32), *t1_m2w=IN(33), *t1_m2b=IN(34);
    // gmpool
    const float *gm_l1w=IN(35), *gm_l1b=IN(36), *gm_n1g=IN(37), *gm_n1b=IN(38);
    const float *gm_l2w=IN(39), *gm_l2b=IN(40);
    // garo.mlp
    const float *ga_l1w=IN(41), *ga_l1b=IN(42), *ga_n1g=IN(43), *ga_n1b=IN(44);
    const float *ga_l2w=IN(45), *ga_l2b=IN(46), *ga_n2g=IN(47), *ga_n2b=IN(48);
    // garo.mlp2
    const float *g2_l1w=IN(49), *g2_l1b=IN(50), *g2_n1g=IN(51), *g2_n1b=IN(52);
    const float *g2_l2w=IN(53), *g2_l2b=IN(54), *g2_n2g=IN(55), *g2_n2b=IN(56);
    const float *qw=IN(57), *qb=IN(58), *kw=IN(59), *kb=IN(60);
    // tatt2
    const float *t2_inw=IN(61), *t2_inb=IN(62), *t2_outw=IN(63), *t2_outb=IN(64);
    const float *t2_ln1g=IN(65), *t2_ln1b=IN(66), *t2_ln2g=IN(67), *t2_ln2b=IN(68);
    const float *t2_m1w=IN(69), *t2_m1b=IN(70), *t2_m2w=IN(71), *t2_m2b=IN(72);
    const float *ow=IN(73), *ob=IN(74);
    #undef IN

    const int rows = B_ * T_;             // 1408
    const long TT2 = (long)B_ * T_ * T_;  // 247808

    float* W = (float*)d_ws;
    size_t off = 0;
    auto alloc = [&](size_t n) { float* p = W + off; off += n; return p; };

    float* mv      = alloc(1152);            // bn stats (2*512 max)
    float* es_tmp  = alloc(B_ * H_);
    float* e_s     = alloc(B_ * H_);
    float* hcat    = alloc((size_t)rows * (T_ + IND_));
    float* ra      = alloc((size_t)rows * 256);
    float* rb      = alloc((size_t)rows * 256);
    float* rc      = alloc((size_t)rows * 256);
    float* rd      = alloc((size_t)rows * 256);
    float* xW      = alloc((size_t)rows * 4 * H_);
    float* x_lstm  = alloc((size_t)rows * H_);
    float* qkv     = alloc((size_t)rows * 3 * T_);   // covers 3*176 and 3*128
    float* scores  = alloc((size_t)B_ * NH_ * T_ * T_);
    float* xe      = alloc((size_t)rows * H_);
    float* big     = alloc((size_t)TT2 * H_);        // dist activations, later corr
    float* clf     = alloc((size_t)TT2);
    float* Amat    = alloc((size_t)TT2);
    float* Vmat    = alloc((size_t)TT2);
    float* Smat    = alloc((size_t)TT2);
    float* fcseq   = alloc((size_t)rows * H_);
    float* hcat2   = alloc((size_t)rows * 2 * H_);
    float* Hn      = alloc((size_t)rows * H_);
    float* hnmean  = alloc(B_ * H_);
    float* xq      = alloc(B_ * H_);
    float* xk      = alloc((size_t)rows * H_);
    float* ev      = alloc((size_t)rows * H_);
    float* xt2     = alloc((size_t)rows * T_);
    float* ha      = alloc((size_t)rows * T_);
    float* hdyn    = alloc(B_ * T_);

    // 1) e_s = mlp2(fc)
    run_mlp2(stream, FC, B_, IND_ * IND_, H_, se_l1w, se_l1b, se_n1g, se_n1b,
             se_l2w, se_l2b, se_n2g, se_n2b, es_tmp, e_s, mv);

    // 2) h = concat(v,t); x = mlp2(h)  (rows b*T+t)
    {
        long tot = (long)rows * (T_ + IND_);
        concat_vt<<<(unsigned)((tot + 255) / 256), 256, 0, stream>>>(V, Tm, hcat, rows, IND_, T_);
    }
    run_mlp2(stream, hcat, rows, T_ + IND_, H_, mp_l1w, mp_l1b, mp_n1g, mp_n1b,
             mp_l2w, mp_l2b, mp_n2g, mp_n2b, ra, rb, mv);

    // 3) LSTM: xW = x@Wih^T + bih; scan adds bhh
    launch_gemm(stream, rb, wih, bih, xW, rows, 4 * H_, H_, H_, H_, 4 * H_, 1, 0, 1.f);
    lstm_kernel<<<1, 512, 0, stream>>>(xW, whh, bhh, x_lstm, T_, B_, H_);

    // 4) transformer tatt (E=128, hid=256) -> xe (rows l*B+b)
    run_transformer(stream, x_lstm, H_, 2 * H_, NH_,
                    t1_inw, t1_inb, t1_outw, t1_outb, t1_ln1g, t1_ln1b, t1_ln2g, t1_ln2b,
                    t1_m1w, t1_m1b, t1_m2w, t1_m2b,
                    qkv, scores, ra, rb, rc, rd, xe);

    // 5) gmpool: dist-lin + BN + relu + l2(sigmoid) -> clf
    {
        dim3 grid((H_ + 127) / 128, (unsigned)((TT2 + 15) / 16));
        dist_gemm_wmma<<<grid, 128, 0, stream>>>(xe, gm_l1w, gm_l1b, big, T_, B_, H_);
        run_bn(stream, big, mv, TT2, H_, gm_n1g, gm_n1b, 1);
        launch_gemm(stream, big, gm_l2w, gm_l2b, clf, (int)TT2, 1, H_, H_, H_, 1, 1, 2, 1.f);
    }

    // 6) M matrix, eigendecomposition, S = vecs*sqrt(vals) (descending)
    build_M_initV<<<(unsigned)((TT2 + 255) / 256), 256, 0, stream>>>(clf, Amat, Vmat, T_, B_);
    jacobi_kernel<<<B_, 256, 0, stream>>>(Amat, Vmat, T_, 8);
    eigsort_scale<<<B_, 256, 0, stream>>>(Amat, Vmat, Smat, T_);

    // 7) weighted corrcoef per (b,t) -> big (reused), then garo.mlp -> fcseq
    corr_wmma<<<rows, 256, 0, stream>>>(xe, Smat, big, T_, B_, H_);
    corr_normalize<<<rows, 256, 0, stream>>>(big, H_);
    run_mlp2(stream, big, rows, H_ * H_, H_, ga_l1w, ga_l1b, ga_n1g, ga_n1b,
             ga_l2w, ga_l2b, ga_n2g, ga_n2b, ra, fcseq, mv);

    // 8) Hcat = [fcseq, e_s bcast]; Hn = mlp2(Hcat)
    {
        long tot = (long)rows * 2 * H_;
        concat_es<<<(unsigned)((tot + 255) / 256), 256, 0, stream>>>(fcseq, e_s, hcat2, T_, H_);
    }
    run_mlp2(stream, hcat2, rows, 2 * H_, H_, g2_l1w, g2_l1b, g2_n1g, g2_n1b,
             g2_l2w, g2_l2b, g2_n2g, g2_n2b, ra, Hn, mv);

    // 9) garo attention gate: xq, xk, att, ev
    mean_over_t<<<B_, H_, 0, stream>>>(Hn, hnmean, T_, H_);
    launch_gemm(stream, hnmean, qw, qb, xq, B_, H_, H_, H_, H_, H_, 1, 0, 1.f);
    launch_gemm(stream, Hn, kw, kb, xk, rows, H_, H_, H_, H_, H_, 1, 0, 1.f);
    garo_att_ev<<<B_ * T_, H_, 0, stream>>>(xq, xk, Hn, ev, T_, H_);

    // 10) Xres[b,t,s] = sum_h Xe[b,t,h]*ev[b,s,h]  -> xt2 layout (t*B+b, s)
    launch_gemm(stream, xe, ev, nullptr, xt2, T_, T_, H_,
                B_ * H_, H_, B_ * T_, 1, 0, 1.f,
                B_, 1, H_, 0, (long)T_ * H_, 0, T_, 0);

    // 11) transformer tatt2 (E=176, hid=128) -> ha
    run_transformer(stream, xt2, T_, H_, NH_,
                    t2_inw, t2_inb, t2_outw, t2_outb, t2_ln1g, t2_ln1b, t2_ln2g, t2_ln2b,
                    t2_m1w, t2_m1b, t2_m2w, t2_m2b,
                    qkv, scores, ra, rb, rc, rd, ha);

    // 12) h_dyn = sum over L; final linear -> d_out (8 x 2)
    sum_over_l<<<B_, 192, 0, stream>>>(ha, hdyn, T_, B_, T_);
    launch_gemm(stream, hdyn, ow, ob, (float*)d_out, B_, 2, T_, T_, T_, 2, 1, 0, 1.f);
}